// Voxelization_59820304498936
// MI455X (gfx1250) — compile-verified
//
#include <hip/hip_runtime.h>
#include <cstdint>

// Dynamic voxelization, MI455X (gfx1250).
// Memory-bound: 224 MB total traffic, ~9.6 us floor @ 23.3 TB/s.
// Strategy: b128 NT loads (1 per point), b128 NT stores (4 points packed per
// output plane), wave32 blocks, 4 points per thread for MLP.

typedef float  __attribute__((ext_vector_type(4))) f32x4;
typedef int    __attribute__((ext_vector_type(4))) i32x4;

// Grid constants (from reference): voxel (0.05, 0.05, 0.1),
// range (0,-40,-3)..(70.4,40,1), grid (1408, 1600, 40).
__device__ __forceinline__ void voxel_coord(float x, float y, float z,
                                            int& cx, int& cy, int& cz) {
    // Match reference exactly: floor((p - mn) / vs) with f32 division, then i32 cvt.
    float fx = floorf((x - 0.0f)   / 0.05f);
    float fy = floorf((y + 40.0f)  / 0.05f);
    float fz = floorf((z + 3.0f)   / 0.1f);
    int ix = (int)fx;
    int iy = (int)fy;
    int iz = (int)fz;
    bool valid = (ix >= 0) & (ix < 1408) &
                 (iy >= 0) & (iy < 1600) &
                 (iz >= 0) & (iz < 40);
    cx = valid ? ix : -1;
    cy = valid ? iy : -1;
    cz = valid ? iz : -1;
}

// Vector path: each thread handles 4 consecutive points.
// Loads:  4 x global_load_b128 (nt)      = 64 B
// Stores: 3 x global_store_b128 (nt)     = 48 B
__global__ __launch_bounds__(256) void voxelize_vec4(const f32x4* __restrict__ pts,
                                                     int* __restrict__ out,
                                                     int nvec, int n) {
    int t = blockIdx.x * blockDim.x + threadIdx.x;
    if (t >= nvec) return;
    const int base = t * 4;

    i32x4 cx, cy, cz;
#pragma unroll
    for (int k = 0; k < 4; ++k) {
        f32x4 p = __builtin_nontemporal_load(&pts[base + k]);
        int ix, iy, iz;
        voxel_coord(p.x, p.y, p.z, ix, iy, iz);
        cx[k] = ix; cy[k] = iy; cz[k] = iz;
    }

    // (3, N) planes; base is a multiple of 4 -> 16B-aligned stores.
    __builtin_nontemporal_store(cx, (i32x4*)(out + 0 * (size_t)n + base));
    __builtin_nontemporal_store(cy, (i32x4*)(out + 1 * (size_t)n + base));
    __builtin_nontemporal_store(cz, (i32x4*)(out + 2 * (size_t)n + base));
}

// Scalar tail for n % 4 != 0 (not hit for N = 8M, kept for generality).
__global__ __launch_bounds__(128) void voxelize_tail(const float* __restrict__ pts,
                                                     int* __restrict__ out,
                                                     int start, int n) {
    int i = start + blockIdx.x * blockDim.x + threadIdx.x;
    if (i >= n) return;
    float x = pts[i * 4 + 0];
    float y = pts[i * 4 + 1];
    float z = pts[i * 4 + 2];
    int cx, cy, cz;
    voxel_coord(x, y, z, cx, cy, cz);
    out[0 * (size_t)n + i] = cx;
    out[1 * (size_t)n + i] = cy;
    out[2 * (size_t)n + i] = cz;
}

extern "C" void kernel_launch(void* const* d_in, const int* in_sizes, int n_in,
                              void* d_out, int out_size, void* d_ws, size_t ws_size,
                              hipStream_t stream) {
    const float* pts = (const float*)d_in[0];
    const int n = in_sizes[0] / 4;          // points are (N, 4) f32
    int* out = (int*)d_out;                 // (3, N) int32 bit patterns

    const int nvec = n / 4;
    const int tail = n - nvec * 4;

    if (nvec > 0) {
        const int block = 256;              // 8 wave32 waves
        const int grid  = (nvec + block - 1) / block;
        voxelize_vec4<<<grid, block, 0, stream>>>((const f32x4*)pts, out, nvec, n);
    }
    if (tail > 0) {
        const int block = 128;
        const int grid  = (tail + block - 1) / block;
        voxelize_tail<<<grid, block, 0, stream>>>(pts, out, nvec * 4, n);
    }
}